// GCN_80376017977911
// MI455X (gfx1250) — compile-verified
//
#include <hip/hip_runtime.h>

typedef __attribute__((ext_vector_type(2))) float v2f;
typedef __attribute__((ext_vector_type(4))) float v4f;
typedef __attribute__((ext_vector_type(8))) float v8f;

#define F_IN   256
#define HIDDEN 64

// ---------------- degree / normalization ----------------

__global__ void deg_init_k(float* __restrict__ deg, int n) {
    int i = blockIdx.x * blockDim.x + threadIdx.x;
    if (i < n) deg[i] = 1.0f;                    // self-loop contributes 1
}

__global__ void deg_accum_k(const int* __restrict__ dst, float* deg, int E) {
    int i = blockIdx.x * blockDim.x + threadIdx.x;
    if (i < E) atomicAdd(&deg[dst[i]], 1.0f);
}

__global__ void deg_rsqrt_k(float* deg, int n) {
    int i = blockIdx.x * blockDim.x + threadIdx.x;
    if (i < n) deg[i] = rsqrtf(deg[i]);          // deg >= 1 always (self-loops)
}

// ---------------- fp32 WMMA GEMM: Out[M,Nout] = A[M,K] @ W[K,Nout] ----------------
// Block = 128 threads (4 waves). Block covers 16 rows; wave w covers cols [16w,16w+16).
// V_WMMA_F32_16X16X4_F32: exact fp32 math, matches reference numerics.

template <int K, int NOUT>
__global__ __launch_bounds__(128) void gemm_wmma_k(const float* __restrict__ A,
                                                   const float* __restrict__ W,
                                                   float* __restrict__ Out,
                                                   int M) {
    constexpr int LDSK = K + 4;       // pad 4: rows stay 16B-aligned (b128) and banks decollide
    __shared__ float Atile[16 * LDSK];

    const int tid  = threadIdx.x;
    const int row0 = blockIdx.x * 16;

    // Cooperative float4 (b128) staging of the 16 x K A-tile into LDS.
    constexpr int KV = K / 4;                        // float4s per row
    for (int idx = tid; idx < 16 * KV; idx += 128) {
        int r = idx / KV, c4 = idx % KV;             // powers of two -> shifts
        int row = row0 + r; if (row >= M) row = M - 1;
        v4f v = *(const v4f*)&A[(size_t)row * K + c4 * 4];
        *(v4f*)&Atile[r * LDSK + c4 * 4] = v;
    }
    __syncthreads();

    const int lane = tid & 31;
    const int wave = tid >> 5;
    const int half = lane >> 4;                  // 0: K={k,k+1}, 1: K={k+2,k+3}
    const int l    = lane & 15;
    const int col0 = wave * 16;

    v8f acc = {};
    const float* wp = W + col0 + l;              // column slice of W
    for (int k = 0; k < K; k += 4) {
        const int ka = k + half * 2;
        // A-operand: lane l holds row M=l, two consecutive K values (ISA 16x4 f32 layout)
        v2f a = *(const v2f*)&Atile[l * LDSK + ka];
        // B-operand: rows K=ka, ka+1 striped over lanes (VGPR0/VGPR1 per ISA layout)
        v2f b;
        b.x = wp[ka * NOUT];
        b.y = wp[(ka + 1) * NOUT];
        acc = __builtin_amdgcn_wmma_f32_16x16x4_f32(false, a, false, b,
                                                    (short)0, acc, false, false);
    }

    // C/D layout: VGPR r -> M = r + 8*half, N = l
    float* op = Out + (size_t)(row0 + half * 8) * NOUT + col0 + l;
    if (row0 + 16 <= M) {                        // fast path: full tile, no guards
#pragma unroll
        for (int r = 0; r < 8; ++r) op[r * NOUT] = acc[r];
    } else {                                     // tail tile (unused for N=100000)
#pragma unroll
        for (int r = 0; r < 8; ++r)
            if (row0 + half * 8 + r < M) op[r * NOUT] = acc[r];
    }
}

// ---------------- aggregation ----------------

// Self-loop term: agg[i,f] = h[i,f] * dinv[i]^2   (also serves as the agg init)
template <int F>
__global__ void init_agg_k(const float* __restrict__ h, const float* __restrict__ dinv,
                           float* __restrict__ agg, int n) {
    long long idx = (long long)blockIdx.x * blockDim.x + threadIdx.x;
    if (idx >= (long long)n * F) return;
    int i = (int)(idx / F);
    float d = dinv[i];
    agg[idx] = h[idx] * d * d;
}

// Edge term: agg[dst,f] += h[src,f] * dinv[src]*dinv[dst]
template <int F>
__global__ void edge_agg_k(const int* __restrict__ src, const int* __restrict__ dst,
                           const float* __restrict__ dinv, const float* __restrict__ h,
                           float* agg, int E) {
    long long idx = (long long)blockIdx.x * blockDim.x + threadIdx.x;
    if (idx >= (long long)E * F) return;
    int e = (int)(idx / F), f = (int)(idx % F);
    int s = src[e], d = dst[e];
    float w = dinv[s] * dinv[d];
    atomicAdd(&agg[(size_t)d * F + f], h[(size_t)s * F + f] * w);
}

__global__ void bias_relu_k(float* a, const float* __restrict__ b, long long total) {
    long long idx = (long long)blockIdx.x * blockDim.x + threadIdx.x;
    if (idx >= total) return;
    float v = a[idx] + b[idx & (HIDDEN - 1)];
    a[idx] = v > 0.0f ? v : 0.0f;
}

// ---------------- layer 3 (64 -> 2) fused with self-loop init ----------------

__global__ void gemm3_k(const float* __restrict__ h, const float* __restrict__ W3,
                        const float* __restrict__ dinv, float* __restrict__ h3,
                        float* __restrict__ out, int n) {
    int i = blockIdx.x * blockDim.x + threadIdx.x;
    if (i >= n) return;
    const float* hr = h + (size_t)i * HIDDEN;
    float s0 = 0.0f, s1 = 0.0f;
#pragma unroll
    for (int f = 0; f < HIDDEN; ++f) {
        float v = hr[f];
        s0 += v * W3[2 * f];
        s1 += v * W3[2 * f + 1];
    }
    h3[2 * i] = s0; h3[2 * i + 1] = s1;
    float d2 = dinv[i] * dinv[i];
    out[2 * i] = s0 * d2; out[2 * i + 1] = s1 * d2;   // self-loop init of agg3
}

__global__ void logsoftmax_k(float* out, const float* __restrict__ b3, int n) {
    int i = blockIdx.x * blockDim.x + threadIdx.x;
    if (i >= n) return;
    float z0 = out[2 * i] + b3[0];
    float z1 = out[2 * i + 1] + b3[1];
    float m = fmaxf(z0, z1);
    float lse = m + logf(__expf(z0 - m) + __expf(z1 - m));
    out[2 * i] = z0 - lse;
    out[2 * i + 1] = z1 - lse;
}

// ---------------- launcher ----------------

extern "C" void kernel_launch(void* const* d_in, const int* in_sizes, int n_in,
                              void* d_out, int out_size, void* d_ws, size_t ws_size,
                              hipStream_t stream) {
    const float* x  = (const float*)d_in[0];
    const int*   ei = (const int*)d_in[1];
    const float* W1 = (const float*)d_in[2];
    const float* b1 = (const float*)d_in[3];
    const float* W2 = (const float*)d_in[4];
    const float* b2 = (const float*)d_in[5];
    const float* W3 = (const float*)d_in[6];
    const float* b3 = (const float*)d_in[7];
    float* out = (float*)d_out;

    const int N = in_sizes[0] / F_IN;       // 100000
    const int E = in_sizes[1] / 2;          // 1600000
    const int* src = ei;
    const int* dst = ei + E;

    float* ws   = (float*)d_ws;             // layout: dinv | bufA | bufB | h3
    float* dinv = ws;
    float* bufA = dinv + N;
    float* bufB = bufA + (size_t)N * HIDDEN;
    float* h3   = bufB + (size_t)N * HIDDEN;

    const int T = 256;
    dim3 blk(T);
    int gN   = (N + T - 1) / T;
    int gE   = (E + T - 1) / T;
    long long nh = (long long)N * HIDDEN;
    long long eh = (long long)E * HIDDEN;
    int gNH  = (int)((nh + T - 1) / T);
    int gEH  = (int)((eh + T - 1) / T);
    int gE2  = (int)(((long long)E * 2 + T - 1) / T);
    int gTile = (N + 15) / 16;

    // normalization: deg -> dinv = rsqrt(deg)
    deg_init_k<<<gN, blk, 0, stream>>>(dinv, N);
    deg_accum_k<<<gE, blk, 0, stream>>>(dst, dinv, E);
    deg_rsqrt_k<<<gN, blk, 0, stream>>>(dinv, N);

    // layer 1: h = x @ W1 (WMMA fp32), aggregate, bias+relu
    gemm_wmma_k<F_IN, HIDDEN><<<gTile, dim3(128), 0, stream>>>(x, W1, bufA, N);
    init_agg_k<HIDDEN><<<gNH, blk, 0, stream>>>(bufA, dinv, bufB, N);
    edge_agg_k<HIDDEN><<<gEH, blk, 0, stream>>>(src, dst, dinv, bufA, bufB, E);
    bias_relu_k<<<gNH, blk, 0, stream>>>(bufB, b1, nh);

    // layer 2
    gemm_wmma_k<HIDDEN, HIDDEN><<<gTile, dim3(128), 0, stream>>>(bufB, W2, bufA, N);
    init_agg_k<HIDDEN><<<gNH, blk, 0, stream>>>(bufA, dinv, bufB, N);
    edge_agg_k<HIDDEN><<<gEH, blk, 0, stream>>>(src, dst, dinv, bufA, bufB, E);
    bias_relu_k<<<gNH, blk, 0, stream>>>(bufB, b2, nh);

    // layer 3: 64 -> 2, aggregate into d_out, log_softmax in place
    gemm3_k<<<gN, blk, 0, stream>>>(bufB, W3, dinv, h3, out, N);
    edge_agg_k<2><<<gE2, blk, 0, stream>>>(src, dst, dinv, h3, out, E);
    logsoftmax_k<<<gN, blk, 0, stream>>>(out, b3, N);
}